// GNNLinear_35210141892662
// MI455X (gfx1250) — compile-verified
//
#include <hip/hip_runtime.h>
#include <hip/hip_bf16.h>
#include <math.h>

// GCN: 2x (linear via WMMA f32_16x16x4 + sym-normalized scatter-add), then log_softmax.
// N=100000, F_IN=128, F_HID=64, F_OUT=16, E=1600000.

#define F_IN  128
#define F_HID 64
#define F_OUT 16

typedef float v2f __attribute__((ext_vector_type(2)));
typedef float v8f __attribute__((ext_vector_type(8)));

// gfx1250 async global->LDS DMA path (ASYNCcnt), with safe fallback.
#if defined(__gfx1250__) && __has_builtin(__builtin_amdgcn_global_load_async_to_lds_b128) && \
    __has_builtin(__builtin_amdgcn_s_wait_asynccnt)
#define GCN_ASYNC_LDS 1
#else
#define GCN_ASYNC_LDS 0
#endif

__device__ __forceinline__ void block_fill_lds(float* dst, const float* __restrict__ src,
                                               int nfloats) {
#if GCN_ASYNC_LDS
  // Builtin signature (from clang diagnostic): (v4i __device__*, v4i __shared__*, imm, imm)
  typedef int v4i_vs __attribute__((vector_size(16)));
  typedef __attribute__((address_space(1))) v4i_vs* gas_t;
  typedef __attribute__((address_space(3))) v4i_vs* las_t;
  const int nchunks = nfloats >> 2;  // 16B per async op per lane
  for (int i = threadIdx.x; i < nchunks; i += blockDim.x) {
    __builtin_amdgcn_global_load_async_to_lds_b128((gas_t)(src + 4 * i), (las_t)(dst + 4 * i),
                                                   0, 0);
  }
  __builtin_amdgcn_s_wait_asynccnt(0);
#else
  for (int i = threadIdx.x; i < nfloats; i += blockDim.x) dst[i] = src[i];
#endif
  __syncthreads();
}

// ---------------------------------------------------------------- utilities

__global__ void zero4_kernel(float4* __restrict__ p, long count4) {
  long stride = (long)gridDim.x * blockDim.x;
  float4 z = make_float4(0.f, 0.f, 0.f, 0.f);
  for (long i = (long)blockIdx.x * blockDim.x + threadIdx.x; i < count4; i += stride)
    p[i] = z;
}

__global__ void degree_kernel(const int* __restrict__ col, float* __restrict__ deg, int E) {
  int e = blockIdx.x * blockDim.x + threadIdx.x;
  if (e < E) atomicAdd(&deg[col[e]], 1.0f);
}

__global__ void dis_kernel(const float* __restrict__ deg, float* __restrict__ dis, int n) {
  int i = blockIdx.x * blockDim.x + threadIdx.x;
  if (i < n) {
    float d = deg[i];
    dis[i] = (d > 0.0f) ? rsqrtf(fmaxf(d, 1.0f)) : 0.0f;
  }
}

// ---------------------------------------------------------------- GEMM 1: h1 = x @ W1
// One wave computes a 16(M) x 64(N) strip. K=128 -> 32 k-steps of V_WMMA_F32_16X16X4_F32
// over 4 N-tiles = 128 WMMA per wave. W1 (128x64 f32, 32KB) staged in LDS.
__global__ __launch_bounds__(256) void gemm1_wmma(const float* __restrict__ x,
                                                  const float* __restrict__ W1,
                                                  float* __restrict__ h1, int n) {
  __shared__ float w1s[F_IN * F_HID];
  block_fill_lds(w1s, W1, F_IN * F_HID);

  const int lane = threadIdx.x & 31;
  const int wave = threadIdx.x >> 5;
  const int m0 = (blockIdx.x * 8 + wave) * 16;
  if (m0 >= n) return;                       // wave-uniform: EXEC stays all-1s for WMMA

  const int nc = lane & 15;                  // column within 16-wide tile
  const int h  = lane >> 4;                  // K half-select (A/B layout)
  int mrow = m0 + nc;
  const int m = (mrow < n) ? mrow : (n - 1); // clamp tail rows (stores are guarded)

  v8f acc[4] = {};
  const float* xrow = x + (long)m * F_IN + 2 * h;

  for (int k = 0; k < F_IN; k += 4) {
    // A tile 16x4 f32: lane holds M=lane&15, K = 2*h + {0,1} within the k-slab
    v2f a;
    a.x = xrow[k];
    a.y = xrow[k + 1];
    const int kb = k + 2 * h;
#pragma unroll
    for (int j = 0; j < 4; ++j) {
      // B tile 4x16 f32: lane supplies N=lane&15, K = 2*h + {0,1}
      v2f b;
      b.x = w1s[kb * F_HID + j * 16 + nc];
      b.y = w1s[(kb + 1) * F_HID + j * 16 + nc];
      acc[j] = __builtin_amdgcn_wmma_f32_16x16x4_f32(false, a, false, b,
                                                     (short)0, acc[j], false, false);
    }
  }

  // C/D layout: VGPR r -> M = r + 8*h, N = lane&15
  const int mr = h * 8;
#pragma unroll
  for (int j = 0; j < 4; ++j) {
#pragma unroll
    for (int r = 0; r < 8; ++r) {
      int mm = m0 + mr + r;
      if (mm < n) h1[(long)mm * F_HID + j * 16 + nc] = acc[j][r];
    }
  }
}

// ---------------------------------------------------------------- GEMM 2: h2 = a1 @ W2
// One wave computes a 16x16 tile, K=64 -> 16 WMMA per wave. W2 (64x16, 4KB) in LDS.
__global__ __launch_bounds__(256) void gemm2_wmma(const float* __restrict__ a1,
                                                  const float* __restrict__ W2,
                                                  float* __restrict__ h2, int n) {
  __shared__ float w2s[F_HID * F_OUT];
  block_fill_lds(w2s, W2, F_HID * F_OUT);

  const int lane = threadIdx.x & 31;
  const int wave = threadIdx.x >> 5;
  const int m0 = (blockIdx.x * 8 + wave) * 16;
  if (m0 >= n) return;

  const int nc = lane & 15;
  const int h  = lane >> 4;
  int mrow = m0 + nc;
  const int m = (mrow < n) ? mrow : (n - 1);

  v8f acc = {};
  const float* arow = a1 + (long)m * F_HID + 2 * h;

  for (int k = 0; k < F_HID; k += 4) {
    v2f a;
    a.x = arow[k];
    a.y = arow[k + 1];
    const int kb = k + 2 * h;
    v2f b;
    b.x = w2s[kb * F_OUT + nc];
    b.y = w2s[(kb + 1) * F_OUT + nc];
    acc = __builtin_amdgcn_wmma_f32_16x16x4_f32(false, a, false, b,
                                                (short)0, acc, false, false);
  }

  const int mr = h * 8;
#pragma unroll
  for (int r = 0; r < 8; ++r) {
    int mm = m0 + mr + r;
    if (mm < n) h2[(long)mm * F_OUT + nc] = acc[r];
  }
}

// ---------------------------------------------------------------- edge scatter (L2-resident)
// 16 threads/edge; each thread gathers one float4 of h1[row] (global_load_b128) and does
// 4 f32 atomic adds into a1[col]. a1/h1 fit in the 192MB L2, so this is L2 bandwidth/atomic
// limited, not HBM limited.
__global__ void scatter1_kernel(const int* __restrict__ row, const int* __restrict__ col,
                                const float* __restrict__ dis, const float* __restrict__ h1,
                                float* __restrict__ a1, long total) {
  long idx = (long)blockIdx.x * blockDim.x + threadIdx.x;
  if (idx >= total) return;
  int e = (int)(idx >> 4);
  int q = (int)(idx & 15);
  int r = row[e], c = col[e];
  float w = dis[r] * dis[c];
  float4 m = ((const float4*)h1)[(long)r * (F_HID / 4) + q];
  float* dst = a1 + (long)c * F_HID + q * 4;
  atomicAdd(dst + 0, m.x * w);
  atomicAdd(dst + 1, m.y * w);
  atomicAdd(dst + 2, m.z * w);
  atomicAdd(dst + 3, m.w * w);
}

// 4 threads/edge, float4 gather of h2[row], atomic add into out[col].
__global__ void scatter2_kernel(const int* __restrict__ row, const int* __restrict__ col,
                                const float* __restrict__ dis, const float* __restrict__ h2,
                                float* __restrict__ out, long total) {
  long idx = (long)blockIdx.x * blockDim.x + threadIdx.x;
  if (idx >= total) return;
  int e = (int)(idx >> 2);
  int q = (int)(idx & 3);
  int r = row[e], c = col[e];
  float w = dis[r] * dis[c];
  float4 m = ((const float4*)h2)[(long)r * (F_OUT / 4) + q];
  float* dst = out + (long)c * F_OUT + q * 4;
  atomicAdd(dst + 0, m.x * w);
  atomicAdd(dst + 1, m.y * w);
  atomicAdd(dst + 2, m.z * w);
  atomicAdd(dst + 3, m.w * w);
}

// ---------------------------------------------------------------- log_softmax (in place, 16 cols)
__global__ void logsoftmax_kernel(float* __restrict__ out, int n) {
  int i = blockIdx.x * blockDim.x + threadIdx.x;
  if (i >= n) return;
  float4* rp4 = (float4*)(out + (long)i * F_OUT);
  float4 v4[4];
  float v[F_OUT];
#pragma unroll
  for (int q = 0; q < 4; ++q) v4[q] = rp4[q];
#pragma unroll
  for (int q = 0; q < 4; ++q) {
    v[4 * q + 0] = v4[q].x; v[4 * q + 1] = v4[q].y;
    v[4 * q + 2] = v4[q].z; v[4 * q + 3] = v4[q].w;
  }
  float mx = -INFINITY;
#pragma unroll
  for (int f = 0; f < F_OUT; ++f) mx = fmaxf(mx, v[f]);
  float s = 0.0f;
#pragma unroll
  for (int f = 0; f < F_OUT; ++f) s += expf(v[f] - mx);
  float lse = mx + logf(s);
#pragma unroll
  for (int q = 0; q < 4; ++q)
    rp4[q] = make_float4(v[4 * q + 0] - lse, v[4 * q + 1] - lse,
                         v[4 * q + 2] - lse, v[4 * q + 3] - lse);
}

// ---------------------------------------------------------------- launch

extern "C" void kernel_launch(void* const* d_in, const int* in_sizes, int n_in,
                              void* d_out, int out_size, void* d_ws, size_t ws_size,
                              hipStream_t stream) {
  const float* x  = (const float*)d_in[0];
  const int*   ei = (const int*)d_in[1];
  const float* W1 = (const float*)d_in[2];
  const float* W2 = (const float*)d_in[3];
  float* out = (float*)d_out;

  const int n = in_sizes[0] / F_IN;   // 100000
  const int E = in_sizes[1] / 2;      // 1600000
  const int* rowi = ei;               // edge_index[0] = source
  const int* coli = ei + E;           // edge_index[1] = target

  // workspace layout (floats): deg[n] | dis[n] | h1[n*64] | a1[n*64] | h2[n*16]
  float* ws  = (float*)d_ws;
  float* deg = ws;
  float* dis = ws + n;
  float* h1  = ws + 2L * n;
  float* a1  = h1 + (long)n * F_HID;
  float* h2  = a1 + (long)n * F_HID;

  const int T = 256;

  zero4_kernel<<<512, T, 0, stream>>>((float4*)deg, (long)n / 4);
  zero4_kernel<<<2048, T, 0, stream>>>((float4*)a1, (long)n * F_HID / 4);
  zero4_kernel<<<1024, T, 0, stream>>>((float4*)out, (long)n * F_OUT / 4);

  degree_kernel<<<(E + T - 1) / T, T, 0, stream>>>(coli, deg, E);
  dis_kernel<<<(n + T - 1) / T, T, 0, stream>>>(deg, dis, n);

  // layer 1
  gemm1_wmma<<<(n + 127) / 128, T, 0, stream>>>(x, W1, h1, n);
  long tot1 = (long)E * (F_HID / 4);  // 16 threads per edge
  scatter1_kernel<<<(int)((tot1 + T - 1) / T), T, 0, stream>>>(rowi, coli, dis, h1, a1, tot1);

  // layer 2
  gemm2_wmma<<<(n + 127) / 128, T, 0, stream>>>(a1, W2, h2, n);
  long tot2 = (long)E * (F_OUT / 4);  // 4 threads per edge
  scatter2_kernel<<<(int)((tot2 + T - 1) / T), T, 0, stream>>>(rowi, coli, dis, h2, out, tot2);

  logsoftmax_kernel<<<(n + T - 1) / T, T, 0, stream>>>(out, n);
}